// multi_fuse_pic_cls_73212012527799
// MI455X (gfx1250) — compile-verified
//
#include <hip/hip_runtime.h>

typedef __attribute__((ext_vector_type(2))) float v2f;
typedef __attribute__((ext_vector_type(8))) float v8f;
typedef __attribute__((ext_vector_type(4))) unsigned int u32x4;
typedef __attribute__((ext_vector_type(4))) int i32x4;
typedef __attribute__((ext_vector_type(8))) int i32x8;

#define B_SZ 8192
#define D_SZ 2048
#define M_SZ 8
#define C_SZ 1000
#define TB   128                   // b-rows per block
#define KBLK 32                    // k per LDS stage
#define XSTR 36                    // x row stride in LDS (32 data + 4 pad, via TDM pad)
#define BKSTR 18                   // weight k-row stride in LDS (16 data + 2 pad, via TDM pad)
#define NGRP 16                    // 8 model + 8 resnet column groups
#define XS_DW   (TB * XSTR)        // 4608 dwords
#define BG_DW   (KBLK * BKSTR)     // 576 dwords per weight-group tile
#define BS_DW   (NGRP * BG_DW)     // 9216 dwords
#define SBUF_DW (XS_DW + BS_DW)    // 13824 dwords per buffer (55,296 B)
#define NSTAGE  (D_SZ / KBLK)      // 64

// Issue one 2D TDM tile load: tile0 contiguous elements (X), tile1 rows (Y),
// row stride stride0 (elements), with LDS padding for bank-conflict-free strides.
// OOB in X (>= dim0 from tile start) is zero-filled by the TDM.
__device__ __forceinline__ void tdm_load_2d(uint32_t lds_off, uint64_t va,
                                            uint32_t dim0, uint32_t tile0, uint32_t tile1,
                                            uint32_t stride0, uint32_t pad_int, uint32_t pad_amt)
{
    u32x4 g0;
    g0[0] = 1u;                                                   // count=1 (valid, user)
    g0[1] = lds_off;                                              // LDS byte address
    g0[2] = (uint32_t)va;                                         // global_addr[31:0]
    g0[3] = (uint32_t)((va >> 32) & 0x01FFFFFFu) | (2u << 30);    // addr[56:32] | type=2
    i32x8 g1;
    g1[0] = (int)(0x00020000u | (1u << 20) | (pad_int << 22) | (pad_amt << 25)); // 4B, pad
    g1[1] = (int)(dim0 << 16);                 // tensor_dim0[15:0] (abar_addr=0)
    g1[2] = (int)(dim0 >> 16);                 // tensor_dim0[31:16] | tensor_dim1[15:0]=0
    g1[3] = (int)((1u << 14) | (tile0 << 16)); // tensor_dim1=1<<30 (hi16) | tile_dim0
    g1[4] = (int)tile1;                        // tile_dim1 | tile_dim2=0
    g1[5] = (int)stride0;                      // tensor_dim0_stride[31:0]
    g1[6] = 0;                                 // stride0[47:32]=0 | stride1[15:0]=0
    g1[7] = 0;
    i32x4 z4 = {0, 0, 0, 0};
    i32x8 z8 = {0, 0, 0, 0, 0, 0, 0, 0};
    __builtin_amdgcn_tensor_load_to_lds(g0, g1, z4, z4, z8, 0);
}

__global__ __launch_bounds__(256)
void multi_fuse_pic_cls_kernel(const float* __restrict__ x,
                               const float* __restrict__ mw,   // [M, D, C]
                               const float* __restrict__ mb,   // [M, C]
                               const float* __restrict__ rw,   // [D, M*C]
                               const float* __restrict__ rb,   // [M*C]
                               float* __restrict__ out)        // [B, C]
{
    __shared__ __attribute__((aligned(16))) float smem[2 * SBUF_DW];  // 110,592 B

    const int tid  = threadIdx.x;
    const int lane = tid & 31;
    const int wave = __builtin_amdgcn_readfirstlane(tid >> 5);   // scalar wave id
    const int nl   = lane & 15;          // N index within 16x16 frag
    const int kh   = (lane >> 4) << 1;   // k offset: 0 (lanes 0-15) or 2 (lanes 16-31)

    const int b0 = blockIdx.x * TB;
    const int c0 = blockIdx.y * 16;
    const uint32_t dim0c = (uint32_t)(C_SZ - c0);  // valid c columns from tile start

    v8f acc[NGRP];
    #pragma unroll
    for (int g = 0; g < NGRP; ++g)
        acc[g] = v8f{0.f, 0.f, 0.f, 0.f, 0.f, 0.f, 0.f, 0.f};

    const uint32_t lds_base = (uint32_t)(uintptr_t)smem;   // low 32 bits = LDS offset
    const uint64_t xva  = (uint64_t)(uintptr_t)x;
    const uint64_t mwva = (uint64_t)(uintptr_t)mw;
    const uint64_t rwva = (uint64_t)(uintptr_t)rw;

    // ---- per-stage TDM issue: 3 descriptors per wave, all scalar math ----
    auto issue_stage = [&](int buf, int d0) {
        const uint32_t boff = (uint32_t)buf * (uint32_t)(SBUF_DW * 4);
        // x chunk: rows [wave*16, wave*16+16), cols [d0, d0+KBLK); pad 32->36 dwords/row
        tdm_load_2d(lds_base + boff + (uint32_t)(wave * 16 * XSTR * 4),
                    xva + ((uint64_t)(b0 + wave * 16) * D_SZ + (uint64_t)d0) * 4,
                    0x40000000u, KBLK, 16, D_SZ, 4, 3);
        // weight groups 2w, 2w+1: 16 c-cols x 32 d-rows; pad 16->18 dwords/row
        #pragma unroll
        for (int t = 0; t < 2; ++t) {
            int g = 2 * wave + t;
            uint64_t va; uint32_t strd;
            if (g < 8) {
                va   = mwva + (((uint64_t)g * D_SZ + (uint64_t)d0) * C_SZ + c0) * 4;
                strd = C_SZ;
            } else {
                va   = rwva + ((uint64_t)(g - 8) * C_SZ + (uint64_t)d0 * (M_SZ * C_SZ) + c0) * 4;
                strd = M_SZ * C_SZ;
            }
            tdm_load_2d(lds_base + boff + (uint32_t)((XS_DW + g * BG_DW) * 4),
                        va, dim0c, 16, KBLK, strd, 3, 1);
        }
    };

    auto compute_stage = [&](int buf) {
        const float* xsC = smem + (size_t)buf * SBUF_DW;
        const float* bsC = xsC + XS_DW;
        #pragma unroll
        for (int kk = 0; kk < KBLK; kk += 4) {
            v2f a = *(const v2f*)(&xsC[(wave * 16 + nl) * XSTR + kk + kh]);
            #pragma unroll
            for (int g = 0; g < NGRP; ++g) {
                const float* bp = &bsC[g * BG_DW + (kk + kh) * BKSTR + nl];
                v2f bf = { bp[0], bp[BKSTR] };   // B[k][n], B[k+1][n] -> ds_load_2addr_b32
                acc[g] = __builtin_amdgcn_wmma_f32_16x16x4_f32(
                    false, a, false, bf, (short)0, acc[g], false, false);
            }
        }
    };

    // ---- double-buffered pipeline: TDM runs while WMMAs execute ----
    issue_stage(0, 0);
    __builtin_amdgcn_s_wait_tensorcnt(0);
    __syncthreads();
    for (int s = 0; s < NSTAGE; ++s) {
        if (s + 1 < NSTAGE) issue_stage((s + 1) & 1, (s + 1) * KBLK);
        compute_stage(s & 1);                            // 128 WMMAs
        if (s + 1 < NSTAGE) __builtin_amdgcn_s_wait_tensorcnt(0);
        __syncthreads();
    }

    // ---- fused epilogue: res[b,c] = sum_m (O_m + bias_m) * (V_m + rbias_m) ----
    const int  cl   = c0 + nl;
    const bool cval = cl < C_SZ;
    v8f res = v8f{0.f, 0.f, 0.f, 0.f, 0.f, 0.f, 0.f, 0.f};
    #pragma unroll
    for (int m = 0; m < 8; ++m) {
        float bm = cval ? mb[m * C_SZ + cl] : 0.0f;
        float rm = cval ? rb[m * C_SZ + cl] : 0.0f;
        #pragma unroll
        for (int r = 0; r < 8; ++r)
            res[r] += (acc[m][r] + bm) * (acc[m + 8][r] + rm);
    }
    if (cval) {
        int rowb = b0 + wave * 16 + ((lane >> 4) << 3);
        #pragma unroll
        for (int r = 0; r < 8; ++r)
            out[(size_t)(rowb + r) * C_SZ + cl] = res[r];
    }
}

extern "C" void kernel_launch(void* const* d_in, const int* in_sizes, int n_in,
                              void* d_out, int out_size, void* d_ws, size_t ws_size,
                              hipStream_t stream) {
    const float* x  = (const float*)d_in[0];
    const float* mw = (const float*)d_in[1];
    const float* mb = (const float*)d_in[2];
    const float* rw = (const float*)d_in[3];
    const float* rb = (const float*)d_in[4];
    float* out = (float*)d_out;

    dim3 grid(B_SZ / TB, (C_SZ + 15) / 16);   // 64 x 63 blocks
    multi_fuse_pic_cls_kernel<<<grid, 256, 0, stream>>>(x, mw, mb, rw, rb, out);
}